// MappingToContinuous_83854941487235
// MI455X (gfx1250) — compile-verified
//
#include <hip/hip_runtime.h>

// Z[b,i,k] = (k == C[b,i]) ? u : ndtri(clip(U[b,i,k] * Phi(u), 1e-7, 1-1e-7))
// u = MU + SIGMA*eps = eps (MU=0, SIGMA=1), K = 128, rows = B*N = 262144.
//
// One wave32 per 8 rows; K=128 = 32 lanes * float4, lane L covers k=4L..4L+3.
// Row constants via readfirstlane-scalarized 128-bit loads; Phi computed
// lane-parallel once per thread, broadcast with v_readlane_b32.
// ndtri pipeline is CDNA5 packed fp32 (v_pk_*_f32), all-VGPR operands
// (gfx1250 assembler rejects SGPR sources on packed-f32 ops). Coefficient
// pairs are pinned with an empty asm so the RA materializes each exactly
// once (v_mov_b64) and cannot rematerialize per call site.
// The clip is a single v_max_f32 on l = log(1-x^2), using 1-x^2 = 4p(1-p):
// l >= log(4e-7*(1-1e-7)) = -14.7318013 covers both clip sides.
// All U/Z traffic is 128-bit non-temporal (single-use 256 MiB stream;
// HBM roofline ~11.5 us at 23.3 TB/s).

#define KDIM 128
#define ROWS_PER_WAVE 8

typedef __attribute__((ext_vector_type(2))) float v2f;
typedef __attribute__((ext_vector_type(4))) float v4f;

__device__ __forceinline__ v2f splat2(float s) { v2f v; v.x = s; v.y = s; return v; }

// Materialize a splatted pair exactly once: the empty asm makes the def
// opaque, so the RA cannot rematerialize it at each use.
__device__ __forceinline__ v2f pin2(float s) {
  v2f v = splat2(s);
  asm("" : "+v"(v));
  return v;
}

__device__ __forceinline__ float rdlanef(float v, int l) {
  return __uint_as_float(__builtin_amdgcn_readlane(__float_as_uint(v), l));
}

// Front of the pipeline, fully packed, fed directly by the load pair:
//   p = u*phi ; x = 2p-1 ; m = 1-x^2
__device__ __forceinline__ void pk_front(v2f u01, v2f php, v2f* xo, v2f* mo) {
  v2f x, m;
  asm("v_pk_mul_f32 %0, %2, %3\n\t"                          // p = u*phi
      "v_pk_fma_f32 %0, %0, 2.0, -1.0 op_sel_hi:[1,0,0]\n\t" // x = 2p-1
      "v_pk_mul_f32 %1, %0, %0\n\t"                          // m = x*x
      "v_pk_fma_f32 %1, %1, -1.0, 1.0 op_sel_hi:[1,0,0]"     // m = 1-x^2
      : "=&v"(x), "=&v"(m)
      : "v"(u01), "v"(php));
  *xo = x;
  *mo = m;
}

// 9-coefficient packed Horner (8 fma steps); first step untied so no
// accumulator-init mov is needed. Coefficients are pinned VGPR pairs.
__device__ __forceinline__ v2f horner9(v2f w,
                                       v2f k0, v2f k1, v2f k2, v2f k3, v2f k4,
                                       v2f k5, v2f k6, v2f k7, v2f k8) {
  v2f acc;
  asm("v_pk_fma_f32 %0, %2, %1, %3\n\t"   // acc = k0*w + k1
      "v_pk_fma_f32 %0, %0, %1, %4\n\t"
      "v_pk_fma_f32 %0, %0, %1, %5\n\t"
      "v_pk_fma_f32 %0, %0, %1, %6\n\t"
      "v_pk_fma_f32 %0, %0, %1, %7\n\t"
      "v_pk_fma_f32 %0, %0, %1, %8\n\t"
      "v_pk_fma_f32 %0, %0, %1, %9\n\t"
      "v_pk_fma_f32 %0, %0, %1, %10"
      : "=&v"(acc)
      : "v"(w), "v"(k0), "v"(k1), "v"(k2), "v"(k3), "v"(k4),
        "v"(k5), "v"(k6), "v"(k7), "v"(k8));
  return acc;
}

// Pinned coefficient pairs (sqrt(2) folded into both Giles erfinv sets).
struct Coeffs {
  v2f c0, c1, c2, c3, c4, c5, c6, c7, c8;   // central branch (w < 5)
  v2f t0, t1, t2, t3, t4, t5, t6, t7, t8;   // tail branch   (w >= 5)
};

// ---- Phi(u) = 0.5*(1 + erf(u/sqrt(2))) , A&S 7.1.26 (|err| <= 1.5e-7) ----
__device__ __forceinline__ float norm_cdf_f(float u) {
  const float x  = u * 0.70710678118654752440f;
  const float ax = __builtin_fabsf(x);
  const float t  = __frcp_rn(__builtin_fmaf(0.3275911f, ax, 1.0f));
  float poly =                        1.061405429f;
  poly = __builtin_fmaf(poly, t, -1.453152027f);
  poly = __builtin_fmaf(poly, t,  1.421413741f);
  poly = __builtin_fmaf(poly, t, -0.284496736f);
  poly = __builtin_fmaf(poly, t,  0.254829592f);
  poly *= t;
  const float e      = __expf(-ax * ax);                 // v_exp_f32
  const float erf_ax = __builtin_fmaf(-poly, e, 1.0f);
  const float erfx   = (x < 0.0f) ? -erf_ax : erf_ax;
  return __builtin_fmaf(0.5f, erfx, 0.5f);
}

// ---- z = ndtri(clip(u*phi)) for a load-aligned pair (Giles erfinv) -------
// Select on l: w<5 <=> l>-5; sqrt(w)=sqrt(-l) folds the negate into the
// v_sqrt source modifier.
__device__ __forceinline__ v2f z_pair(v2f u01, v2f php, const Coeffs& K) {
  v2f x, m;
  pk_front(u01, php, &x, &m);
  // l = max(log(m), log(4e-7*(1-1e-7))) implements the p-clip on both sides.
  const float lx = __builtin_fmaxf(__logf(m.x), -14.7318013f);  // v_log_f32
  const float ly = __builtin_fmaxf(__logf(m.y), -14.7318013f);
  // central branch (w < 5): wc = -l - 2.5
  v2f wc;
  wc.x = __builtin_fmaf(lx, -1.0f, -2.5f);
  wc.y = __builtin_fmaf(ly, -1.0f, -2.5f);
  const v2f pc = horner9(wc, K.c0, K.c1, K.c2, K.c3, K.c4,
                             K.c5, K.c6, K.c7, K.c8);
  // tail branch (w >= 5): ws = sqrt(-l) - 3
  v2f ws;
  ws.x = __fsqrt_rn(-lx) - 3.0f;                                // v_sqrt_f32
  ws.y = __fsqrt_rn(-ly) - 3.0f;
  const v2f ps = horner9(ws, K.t0, K.t1, K.t2, K.t3, K.t4,
                             K.t5, K.t6, K.t7, K.t8);
  v2f z;
  z.x = ((lx > -5.0f) ? pc.x : ps.x) * x.x;
  z.y = ((ly > -5.0f) ? pc.y : ps.y) * x.y;
  return z;
}

__device__ __forceinline__ v4f row_quad(v4f uu, float u, float phi,
                                        int c, int k0, const Coeffs& K) {
  v2f php; php.x = phi; php.y = phi;   // 2 v_mov from SGPR, once per row
  v2f a; a.x = uu.x; a.y = uu.y;       // aligned pairs straight from the load
  v2f b; b.x = uu.z; b.y = uu.w;
  const v2f za = z_pair(a, php, K);
  const v2f zb = z_pair(b, php, K);
  v4f o;
  o.x = (k0 + 0 == c) ? u : za.x;
  o.y = (k0 + 1 == c) ? u : za.y;
  o.z = (k0 + 2 == c) ? u : zb.x;
  o.w = (k0 + 3 == c) ? u : zb.y;
  return o;
}

__global__ __launch_bounds__(256)
void trunc_norm_scatter_kernel(const int* __restrict__ C,
                               const float* __restrict__ eps,
                               const v4f* __restrict__ U4,
                               v4f* __restrict__ Z4,
                               int nrows) {       // nrows = B*N, multiple of 8
  const int lane = threadIdx.x & 31;
  const int wave = (blockIdx.x * blockDim.x + threadIdx.x) >> 5;
  const int R0v  = wave * ROWS_PER_WAVE;
  if (R0v >= nrows) return;                       // wave-uniform exit
  const int R0 = __builtin_amdgcn_readfirstlane(R0v);

  // Pin all polynomial coefficient pairs once (18 x v_mov_b64, no remat).
  Coeffs K;
  K.c0 = pin2(3.97426022e-08f);  K.c1 = pin2( 4.85462660e-07f);
  K.c2 = pin2(-4.98282267e-06f); K.c3 = pin2(-6.21052812e-06f);
  K.c4 = pin2( 3.09120030e-04f); K.c5 = pin2(-1.77303494e-03f);
  K.c6 = pin2(-5.90813406e-03f); K.c7 = pin2( 3.48802662e-01f);
  K.c8 = pin2( 2.12331354e+00f);
  K.t0 = pin2(-2.83145718e-04f); K.t1 = pin2( 1.42765648e-04f);
  K.t2 = pin2( 1.90825948e-03f); K.t3 = pin2(-5.19501231e-03f);
  K.t4 = pin2( 8.11688968e-03f); K.t5 = pin2(-1.07797882e-02f);
  K.t6 = pin2( 1.33485786e-02f); K.t7 = pin2( 1.41658104e+00f);
  K.t8 = pin2( 4.00643425e+00f);

  // Wave-uniform row constants -> scalar (SMEM) 128-bit loads.
  const float4 ea = *(const float4*)(eps + R0);
  const float4 eb = *(const float4*)(eps + R0 + 4);
  const int4   ca = *(const int4*)(C + R0);
  const int4   cb = *(const int4*)(C + R0 + 4);

  // Lane-parallel Phi: lane group j = (lane&7) evaluates Phi(u_j); one
  // norm_cdf per thread covers all 8 rows; broadcast via v_readlane_b32.
  const int sel = lane & 7;
  float uL = ea.x;
  uL = (sel == 1) ? ea.y : uL;
  uL = (sel == 2) ? ea.z : uL;
  uL = (sel == 3) ? ea.w : uL;
  uL = (sel == 4) ? eb.x : uL;
  uL = (sel == 5) ? eb.y : uL;
  uL = (sel == 6) ? eb.z : uL;
  uL = (sel == 7) ? eb.w : uL;
  const float phiL = norm_cdf_f(uL);
  const float phi0 = rdlanef(phiL, 0);
  const float phi1 = rdlanef(phiL, 1);
  const float phi2 = rdlanef(phiL, 2);
  const float phi3 = rdlanef(phiL, 3);
  const float phi4 = rdlanef(phiL, 4);
  const float phi5 = rdlanef(phiL, 5);
  const float phi6 = rdlanef(phiL, 6);
  const float phi7 = rdlanef(phiL, 7);

  const int k0 = lane << 2;
  const long long base = (long long)R0 * (KDIM / 4) + lane;

  // Issue all 8 NT 128-bit loads up front (stream MLP), then compute/store.
  const v4f uu0 = __builtin_nontemporal_load(&U4[base + 0 * 32]);
  const v4f uu1 = __builtin_nontemporal_load(&U4[base + 1 * 32]);
  const v4f uu2 = __builtin_nontemporal_load(&U4[base + 2 * 32]);
  const v4f uu3 = __builtin_nontemporal_load(&U4[base + 3 * 32]);
  const v4f uu4 = __builtin_nontemporal_load(&U4[base + 4 * 32]);
  const v4f uu5 = __builtin_nontemporal_load(&U4[base + 5 * 32]);
  const v4f uu6 = __builtin_nontemporal_load(&U4[base + 6 * 32]);
  const v4f uu7 = __builtin_nontemporal_load(&U4[base + 7 * 32]);

  v4f o;
  o = row_quad(uu0, ea.x, phi0, ca.x, k0, K); __builtin_nontemporal_store(o, &Z4[base + 0 * 32]);
  o = row_quad(uu1, ea.y, phi1, ca.y, k0, K); __builtin_nontemporal_store(o, &Z4[base + 1 * 32]);
  o = row_quad(uu2, ea.z, phi2, ca.z, k0, K); __builtin_nontemporal_store(o, &Z4[base + 2 * 32]);
  o = row_quad(uu3, ea.w, phi3, ca.w, k0, K); __builtin_nontemporal_store(o, &Z4[base + 3 * 32]);
  o = row_quad(uu4, eb.x, phi4, cb.x, k0, K); __builtin_nontemporal_store(o, &Z4[base + 4 * 32]);
  o = row_quad(uu5, eb.y, phi5, cb.y, k0, K); __builtin_nontemporal_store(o, &Z4[base + 5 * 32]);
  o = row_quad(uu6, eb.z, phi6, cb.z, k0, K); __builtin_nontemporal_store(o, &Z4[base + 6 * 32]);
  o = row_quad(uu7, eb.w, phi7, cb.w, k0, K); __builtin_nontemporal_store(o, &Z4[base + 7 * 32]);
}

extern "C" void kernel_launch(void* const* d_in, const int* in_sizes, int n_in,
                              void* d_out, int out_size, void* d_ws, size_t ws_size,
                              hipStream_t stream) {
  // setup_inputs order: C [B,N] int, eps [B,N] f32, U [B,N,K] f32
  const int*   Cp  = (const int*)d_in[0];
  const float* eps = (const float*)d_in[1];
  const v4f*   U4  = (const v4f*)d_in[2];
  v4f*         Z4  = (v4f*)d_out;

  const int nrows   = in_sizes[1];                // B*N = 262144 (mult of 8)
  const int waves   = (nrows + ROWS_PER_WAVE - 1) / ROWS_PER_WAVE;
  const int threads = 256;                        // 8 waves / block
  const int blocks  = (waves * 32 + threads - 1) / threads;

  trunc_norm_scatter_kernel<<<blocks, threads, 0, stream>>>(Cp, eps, U4, Z4, nrows);
}